// Slice_29343216566692
// MI455X (gfx1250) — compile-verified
//
#include <hip/hip_runtime.h>

typedef float v4f __attribute__((ext_vector_type(4)));
typedef __attribute__((address_space(1))) int gas_int;   // global AS
typedef __attribute__((address_space(3))) int las_int;   // LDS AS

#define GN 4
#define GC 12
#define GD 8
#define GH 16
#define GWD 16      // grid width
#define IMH 1024
#define IMW 1024

#define TX 256      // pixels per block in x
#define TY 16       // rows per block in y
#define WX 6        // staged grid-x window
#define WY 3        // staged grid-y window
#define LDS_ELEMS (WY*WX*GD*GC)   // 1728 floats = 6.75 KB

#if __has_builtin(__builtin_amdgcn_global_load_async_to_lds_b32)
#define HAVE_ASYNC_LDS 1
#else
#define HAVE_ASYNC_LDS 0
#endif

__global__ __launch_bounds__(256) void bilateral_slice_kernel(
    const float* __restrict__ grid,   // (N, C, GD, GH, GW)
    const float* __restrict__ guide,  // (N, 1, H, W)
    float* __restrict__ out)          // (N, C, H, W)
{
  __shared__ __align__(16) float lds[LDS_ELEMS];

  const int tid = threadIdx.x;
  const int n   = blockIdx.z;
  const int x0  = blockIdx.x * TX;
  const int y0  = blockIdx.y * TY;

  // Sub-window of the grid needed by this tile (derived analytically).
  int gx0 = x0 >> 6;                       // x0 / 64 image px per grid cell
  int sx  = gx0 - 1;
  sx = sx < 0 ? 0 : (sx > GWD - WX ? GWD - WX : sx);
  int fy_min = (2 * y0 - 63) >> 7;         // floor((y0+0.5)/64 - 0.5)
  int sy = fy_min < 0 ? 0 : (fy_min > GH - WY ? GH - WY : fy_min);

  // ---- Stage sub-grid into LDS, transposed to [ly][lx][lz][c] ----
  const float* gbase = grid + (size_t)n * (GC * GD * GH * GWD);
  for (int i = tid; i < LDS_ELEMS; i += 256) {
    int c  = i % GC;
    int lz = (i / GC) % GD;
    int lx = (i / (GC * GD)) % WX;
    int ly =  i / (GC * GD * WX);
    int gidx = ((c * GD + lz) * GH + (sy + ly)) * GWD + (sx + lx);
#if HAVE_ASYNC_LDS
    __builtin_amdgcn_global_load_async_to_lds_b32(
        (gas_int*)(gbase + gidx),
        (las_int*)&lds[i],
        0, 0);
#else
    lds[i] = gbase[gidx];
#endif
  }
#if HAVE_ASYNC_LDS
#if __has_builtin(__builtin_amdgcn_s_wait_asynccnt)
  __builtin_amdgcn_s_wait_asynccnt(0);
#else
  asm volatile("s_wait_asynccnt 0" ::: "memory");
#endif
#endif
  __syncthreads();

  // ---- Per-thread pixel mapping: 4 consecutive x pixels, 4 row passes ----
  const int gxi = tid & 63;          // x-group within tile (64 groups of 4 px)
  const int x   = x0 + gxi * 4;
  const int r0  = tid >> 6;          // 0..3

  // x weights/indices (independent of row)
  float wx0[4], wx1[4];
  int   lxa[4], lxb[4];
#pragma unroll
  for (int p = 0; p < 4; ++p) {
    float xs = ((float)(x + p) + 0.5f) * (1.0f / 64.0f);   // * GW/W
    float fx = floorf(xs - 0.5f);
    float tx = xs - 0.5f - fx;
    wx0[p] = 1.0f - tx;
    wx1[p] = tx;
    int ix = (int)fx;
    int c0 = ix     < 0 ? 0 : (ix     > GWD - 1 ? GWD - 1 : ix);
    int c1 = ix + 1 < 0 ? 0 : (ix + 1 > GWD - 1 ? GWD - 1 : ix + 1);
    lxa[p] = c0 - sx;
    lxb[p] = c1 - sx;
  }

  const float* gu = guide + (size_t)n * IMH * IMW;
  float*       ob = out   + (size_t)n * GC * IMH * IMW;

#pragma unroll
  for (int k = 0; k < 4; ++k) {
    const int y = y0 + r0 + k * 4;

    float ys = ((float)y + 0.5f) * (1.0f / 64.0f);         // * GH/H
    float fy = floorf(ys - 0.5f);
    float ty = ys - 0.5f - fy;
    float wy0 = 1.0f - ty, wy1 = ty;
    int iy = (int)fy;
    int cy0 = iy     < 0 ? 0 : (iy     > GH - 1 ? GH - 1 : iy);
    int cy1 = iy + 1 < 0 ? 0 : (iy + 1 > GH - 1 ? GH - 1 : iy + 1);
    int lya = cy0 - sy, lyb = cy1 - sy;

    v4f gv = *(const v4f*)(gu + (size_t)y * IMW + x);

    float ch[GC][4];
#pragma unroll
    for (int c = 0; c < GC; ++c)
#pragma unroll
      for (int p = 0; p < 4; ++p) ch[c][p] = 0.0f;

#pragma unroll
    for (int p = 0; p < 4; ++p) {
      float zs = gv[p] * (float)GD;
      float fz = floorf(zs - 0.5f);
      float tz = zs - 0.5f - fz;
      float wzs[2] = { 1.0f - tz, tz };
      int iz = (int)fz;
      int z0 = iz     < 0 ? 0 : (iz     > GD - 1 ? GD - 1 : iz);
      int z1 = iz + 1 < 0 ? 0 : (iz + 1 > GD - 1 ? GD - 1 : iz + 1);
      int lzs[2] = { z0, z1 };

      float wys[2] = { wy0, wy1 };
      int   lys[2] = { lya, lyb };
      float wxs[2] = { wx0[p], wx1[p] };
      int   lxs[2] = { lxa[p], lxb[p] };

#pragma unroll
      for (int a = 0; a < 2; ++a) {
#pragma unroll
        for (int b = 0; b < 2; ++b) {
          float wxy  = wys[a] * wxs[b];
          int   cell = (lys[a] * WX + lxs[b]) * (GD * GC);
#pragma unroll
          for (int d = 0; d < 2; ++d) {
            float w = wxy * wzs[d];
            const float* lp = &lds[cell + lzs[d] * GC];
            v4f l0 = *(const v4f*)(lp);       // ds_load_b128
            v4f l1 = *(const v4f*)(lp + 4);   // ds_load_b128
            v4f l2 = *(const v4f*)(lp + 8);   // ds_load_b128
#pragma unroll
            for (int c = 0; c < 4; ++c) ch[c][p]     += w * l0[c];
#pragma unroll
            for (int c = 0; c < 4; ++c) ch[c + 4][p] += w * l1[c];
#pragma unroll
            for (int c = 0; c < 4; ++c) ch[c + 8][p] += w * l2[c];
          }
        }
      }
    }

    // Streaming (non-temporal) coalesced stores: one b128 per channel.
#pragma unroll
    for (int c = 0; c < GC; ++c) {
      v4f v = { ch[c][0], ch[c][1], ch[c][2], ch[c][3] };
      v4f* dst = (v4f*)(ob + ((size_t)c * IMH + y) * IMW + x);
      __builtin_nontemporal_store(v, dst);
    }
  }
}

extern "C" void kernel_launch(void* const* d_in, const int* in_sizes, int n_in,
                              void* d_out, int out_size, void* d_ws, size_t ws_size,
                              hipStream_t stream) {
  const float* grid  = (const float*)d_in[0];   // (4,12,8,16,16) f32
  const float* guide = (const float*)d_in[1];   // (4,1,1024,1024) f32
  float* out = (float*)d_out;                   // (4,12,1024,1024) f32
  dim3 g(IMW / TX, IMH / TY, GN);               // (4, 64, 4)
  bilateral_slice_kernel<<<g, dim3(256), 0, stream>>>(grid, guide, out);
}